// MaxUnpooling2D_17265768530369
// MI455X (gfx1250) — compile-verified
//
#include <hip/hip_runtime.h>

// MaxUnpooling2D scatter-add, MI455X (gfx1250, wave32).
// B=8, H=256, W=256, C=64, SIZE=(2,2) -> out = [8, 512, 512, 64]
//   batch stride (in)  : H*W*C   = 2^22
//   plane stride (out) : oH*oW*C = 2^24
//   out_flat = (b << 24) | (mask & ~63) | (i & 63)
//
// Memory-bound (0 FLOP/byte): 256 MiB NT streaming reads + 512 MiB zero-fill
// + ~33.5M random 4B L2-atomic RMWs. NT on the streams preserves L2 (192 MB)
// for the atomic working set (~4 atomics/64B output line on average).

#define ELEMS_PER_THREAD 4u
#define BLOCK 256

typedef float __attribute__((ext_vector_type(4))) f32x4;
typedef int   __attribute__((ext_vector_type(4))) i32x4;

// ---------------------------------------------------------------------------
// Fire-and-forget native f32 atomic add (no-return form -> STOREcnt), device
// scope so the RMW executes at L2's atomic units and is visible chip-wide.
// s_endpgm's implicit wait-idle covers the outstanding STOREcnt.
// ---------------------------------------------------------------------------
__device__ __forceinline__ void atomic_add_f32_noret(float* base, unsigned byte_off, float v)
{
    asm volatile("global_atomic_add_f32 %0, %1, %2 scope:SCOPE_DEV"
                 :
                 : "v"(byte_off), "v"(v), "s"(base)
                 : "memory");
}

// ---------------------------------------------------------------------------
// Phase 2: NT streaming b128 reads + 4 no-return global_atomic_add_f32.
// (Defined first so its disassembly leads the .s file.)
// ---------------------------------------------------------------------------
__global__ __launch_bounds__(BLOCK) void unpool_scatter_kernel(
    const f32x4* __restrict__ upd4,
    const i32x4* __restrict__ mask4,
    float*       __restrict__ out)
{
    unsigned tid = blockIdx.x * blockDim.x + threadIdx.x;
    unsigned i0  = tid * ELEMS_PER_THREAD;          // first flat element index

    // read-once streams: th:TH_LOAD_NT keeps them out of the L2 working set
    f32x4 u = __builtin_nontemporal_load(&upd4[tid]);   // global_load_b128 nt
    i32x4 m = __builtin_nontemporal_load(&mask4[tid]);  // global_load_b128 nt

    unsigned base = (i0 >> 22) << 24;               // b * oH*oW*C
    unsigned f    = i0 & 63u;                       // channel of element 0 (4 | C=64)

    // byte offsets into out; all fit in 32 bits (out = 2^29 bytes)
    unsigned o0 = (base | (((unsigned)m.x) & ~63u) | (f + 0u)) << 2;
    unsigned o1 = (base | (((unsigned)m.y) & ~63u) | (f + 1u)) << 2;
    unsigned o2 = (base | (((unsigned)m.z) & ~63u) | (f + 2u)) << 2;
    unsigned o3 = (base | (((unsigned)m.w) & ~63u) | (f + 3u)) << 2;

    atomic_add_f32_noret(out, o0, u.x);
    atomic_add_f32_noret(out, o1, u.y);
    atomic_add_f32_noret(out, o2, u.z);
    atomic_add_f32_noret(out, o3, u.w);
}

// ---------------------------------------------------------------------------
// Phase 1: zero the 512 MiB output with regular-temporal b128 stores so
// freshly zeroed lines are L2-resident when the atomic phase RMWs them.
// ---------------------------------------------------------------------------
__global__ __launch_bounds__(BLOCK) void unpool_zero_kernel(f32x4* __restrict__ out)
{
    size_t i = (size_t)blockIdx.x * blockDim.x + threadIdx.x;
    out[i] = (f32x4){0.f, 0.f, 0.f, 0.f};
}

// ---------------------------------------------------------------------------
extern "C" void kernel_launch(void* const* d_in, const int* in_sizes, int n_in,
                              void* d_out, int out_size, void* d_ws, size_t ws_size,
                              hipStream_t stream)
{
    (void)n_in; (void)d_ws; (void)ws_size;

    const f32x4* updates = (const f32x4*)d_in[0];   // [8,256,256,64] f32
    const i32x4* mask    = (const i32x4*)d_in[1];   // [8,256,256,64] i32
    float*       out     = (float*)d_out;           // [8,512,512,64] f32

    const unsigned n_upd  = (unsigned)in_sizes[0];  // 33,554,432
    const unsigned n_out4 = (unsigned)out_size / 4; // 33,554,432 float4s

    // Phase 1: zero output (stream-ordered before scatter).
    unsigned zblocks = (n_out4 + BLOCK - 1) / BLOCK;
    unpool_zero_kernel<<<zblocks, BLOCK, 0, stream>>>((f32x4*)out);

    // Phase 2: scatter-add, 4 elements per thread.
    unsigned sthreads = n_upd / ELEMS_PER_THREAD;
    unsigned sblocks  = (sthreads + BLOCK - 1) / BLOCK;
    unpool_scatter_kernel<<<sblocks, BLOCK, 0, stream>>>(updates, mask, out);
}